// GCNModel_3702261809849
// MI455X (gfx1250) — compile-verified
//
#include <hip/hip_runtime.h>
#include <hip/hip_bf16.h>

typedef __attribute__((ext_vector_type(2))) float v2f;
typedef __attribute__((ext_vector_type(8))) float v8f;

#define DFEAT 64

// ---------------------------------------------------------------------------
// degree / norm kernels
// ---------------------------------------------------------------------------
__global__ void deg_init_kernel(float* deg, int n) {
    int i = blockIdx.x * blockDim.x + threadIdx.x;
    if (i < n) deg[i] = 1.0f;          // self-loop contributes 1 to every node
}

__global__ void deg_count_kernel(const int* __restrict__ dst, float* deg, int e) {
    int i = blockIdx.x * blockDim.x + threadIdx.x;
    if (i < e) unsafeAtomicAdd(&deg[dst[i]], 1.0f);
}

__global__ void dinv_kernel(float* deg, int n) {   // in place: deg -> deg^{-1/2}
    int i = blockIdx.x * blockDim.x + threadIdx.x;
    if (i < n) deg[i] = rsqrtf(deg[i]);            // deg >= 1 always (ring+self)
}

// ---------------------------------------------------------------------------
// fp32 WMMA GEMM:  C[M,NCOLS] = A[M,K] @ B[K,NCOLS] (+bias, +relu)
// One wave computes a 16-row strip via V_WMMA_F32_16X16X4_F32 (wave32).
// K, NCOLS compile-time: immediate memory offsets, fully unrolled k-loop,
// A strip preloaded into registers and reused across all n-tiles.
// Out-of-range A rows are clamped: they only feed C rows that are never stored.
// ---------------------------------------------------------------------------
template <int K, int NCOLS>
__global__ void gemm_f32_wmma(const float* __restrict__ A, const float* __restrict__ B,
                              const float* __restrict__ bias, float* __restrict__ C,
                              int M, int relu)
{
    int gwave = (blockIdx.x * blockDim.x + threadIdx.x) >> 5;
    int lane  = threadIdx.x & 31;
    int row0  = gwave * 16;
    if (row0 >= M) return;

    int half = lane >> 4;      // 0: lanes 0-15, 1: lanes 16-31
    int l16  = lane & 15;
    int arow = row0 + l16;
    if (arow > M - 1) arow = M - 1;     // clamp (see note above)

    // A 16x4 f32 layout: lane (l16,half) holds A[row][kk], A[row][kk+1],
    // kk = kc*4 + half*2.  Preload all K/4 chunks for this lane.
    constexpr int KC = K / 4;
    v2f a_reg[KC];
    const float* ap = A + arow * K + half * 2;
    #pragma unroll
    for (int kc = 0; kc < KC; ++kc)
        a_reg[kc] = *(const v2f*)(ap + kc * 4);

    // B 4x16 f32 layout: lane holds B[kk][col], B[kk+1][col], col = l16
    const float* bp0 = B + half * 2 * NCOLS + l16;

    #pragma unroll
    for (int nt = 0; nt < NCOLS; nt += 16) {
        v8f acc = {};
        const float* bp = bp0 + nt;
        #pragma unroll
        for (int kc = 0; kc < KC; ++kc) {
            v2f b;
            b.x = bp[kc * 4 * NCOLS];
            b.y = bp[kc * 4 * NCOLS + NCOLS];
            acc = __builtin_amdgcn_wmma_f32_16x16x4_f32(
                /*neg_a=*/false, a_reg[kc], /*neg_b=*/false, b,
                /*c_mod=*/(short)0, acc, /*reuse_a=*/false, /*reuse_b=*/false);
        }
        // C/D 16x16 f32: VGPR i holds row (i + half*8), col = l16
        #pragma unroll
        for (int i = 0; i < 8; ++i) {
            int r = row0 + i + half * 8;
            if (r < M) {
                float v = acc[i];
                if (bias) v += bias[nt + l16];
                if (relu) v = fmaxf(v, 0.0f);
                C[r * NCOLS + nt + l16] = v;
            }
        }
    }
}

// scalar fallback (only used if an unexpected hidden dim shows up)
__global__ void gemm_fallback(const float* __restrict__ A, const float* __restrict__ B,
                              const float* __restrict__ bias, float* __restrict__ C,
                              int M, int K, int Ncols, int relu)
{
    int i = blockIdx.x * blockDim.x + threadIdx.x;
    if (i >= M * Ncols) return;
    int m = i / Ncols, c = i % Ncols;
    float s = bias ? bias[c] : 0.0f;
    for (int k = 0; k < K; ++k) s += A[m * K + k] * B[k * Ncols + c];
    if (relu) s = fmaxf(s, 0.0f);
    C[i] = s;
}

// ---------------------------------------------------------------------------
// edge scatter: acc[dst] += dinv[src]*dinv[dst] * xw[src]
// 2 edges per wave: 16 lanes x float4 = one 64-float feature row per edge.
// ---------------------------------------------------------------------------
__global__ void edge_scatter_kernel(const int* __restrict__ src, const int* __restrict__ dst,
                                    const float* __restrict__ dinv, const float* __restrict__ xw,
                                    float* acc, int e)
{
    int wv   = (blockIdx.x * blockDim.x + threadIdx.x) >> 5;
    int lane = threadIdx.x & 31;
    int edge = wv * 2 + (lane >> 4);
    int l16  = lane & 15;
    if (edge >= e) return;
    int s = src[edge], d = dst[edge];
    float norm = dinv[s] * dinv[d];
    float4 v = *(const float4*)(xw + s * DFEAT + l16 * 4);
    float* p = acc + d * DFEAT + l16 * 4;
    unsafeAtomicAdd(p + 0, v.x * norm);
    unsafeAtomicAdd(p + 1, v.y * norm);
    unsafeAtomicAdd(p + 2, v.z * norm);
    unsafeAtomicAdd(p + 3, v.w * norm);
}

// ---------------------------------------------------------------------------
// finish layer: h = acc + dinv^2 * xw + b  (self-loop folded in), optional relu
// (h may alias acc: purely elementwise)
// ---------------------------------------------------------------------------
__global__ void finish_kernel(const float* acc, const float* __restrict__ xw,
                              const float* __restrict__ dinv, const float* __restrict__ bias,
                              float* h, int n, int relu)
{
    int i = blockIdx.x * blockDim.x + threadIdx.x;
    if (i >= n * DFEAT) return;
    int node = i >> 6;
    int j    = i & (DFEAT - 1);
    float di = dinv[node];
    float v  = acc[i] + di * di * xw[i] + bias[j];
    if (relu) v = fmaxf(v, 0.0f);
    h[i] = v;
}

// ---------------------------------------------------------------------------
// mean pool accumulation: 2 nodes per wave, float4 per lane
// ---------------------------------------------------------------------------
__global__ void pool_accum_kernel(const float* __restrict__ h, const int* __restrict__ batch,
                                  float* sums, float* cnt, int n)
{
    int wv   = (blockIdx.x * blockDim.x + threadIdx.x) >> 5;
    int lane = threadIdx.x & 31;
    int node = wv * 2 + (lane >> 4);
    int l16  = lane & 15;
    if (node >= n) return;
    int g = batch[node];
    float4 v = *(const float4*)(h + node * DFEAT + l16 * 4);
    float* p = sums + g * DFEAT + l16 * 4;
    unsafeAtomicAdd(p + 0, v.x);
    unsafeAtomicAdd(p + 1, v.y);
    unsafeAtomicAdd(p + 2, v.z);
    unsafeAtomicAdd(p + 3, v.w);
    if (l16 == 0) unsafeAtomicAdd(&cnt[g], 1.0f);
}

__global__ void pool_div_kernel(float* sums, const float* __restrict__ cnt, int g)
{
    int i = blockIdx.x * blockDim.x + threadIdx.x;
    if (i >= g * DFEAT) return;
    float c = cnt[i >> 6];
    sums[i] = sums[i] / fmaxf(c, 1.0f);
}

// ---------------------------------------------------------------------------
// final MLP layer: out[g] = sigmoid(z[g,:] . mW2 + mb2)
// ---------------------------------------------------------------------------
__global__ void mlp2_kernel(const float* __restrict__ z, const float* __restrict__ w,
                            const float* __restrict__ b, float* __restrict__ out,
                            int g, int h)
{
    int i = blockIdx.x * blockDim.x + threadIdx.x;
    if (i >= g) return;
    float s = b[0];
    for (int k = 0; k < h; ++k) s += z[i * h + k] * w[k];
    out[i] = 1.0f / (1.0f + expf(-s));
}

// ---------------------------------------------------------------------------
extern "C" void kernel_launch(void* const* d_in, const int* in_sizes, int n_in,
                              void* d_out, int out_size, void* d_ws, size_t ws_size,
                              hipStream_t stream)
{
    const int*   edge_index = (const int*)d_in[0];   // [2, E] row-major
    const int*   batch      = (const int*)d_in[1];   // [N]
    const float* emb        = (const float*)d_in[2]; // [N, 64]
    const float* W1         = (const float*)d_in[3];
    const float* b1         = (const float*)d_in[4];
    const float* W2         = (const float*)d_in[5];
    const float* b2         = (const float*)d_in[6];
    const float* mW1        = (const float*)d_in[7]; // [64, H]
    const float* mb1        = (const float*)d_in[8];
    const float* mW2        = (const float*)d_in[9]; // [H, 1]
    const float* mb2        = (const float*)d_in[10];
    float*       out        = (float*)d_out;

    const int E = in_sizes[0] / 2;
    const int N = in_sizes[1];
    const int G = out_size;
    const int H = in_sizes[7] / DFEAT;      // 128

    const int* src = edge_index;
    const int* dst = edge_index + E;

    // workspace layout (floats)
    float* ws = (float*)d_ws;
    size_t o = 0;
    float* dinv = ws + o; o += (size_t)N;          o = (o + 63) & ~(size_t)63;
    float* bufA = ws + o; o += (size_t)N * DFEAT;  // GEMM output / gather source
    float* bufB = ws + o; o += (size_t)N * DFEAT;  // scatter acc / layer output
    float* sums = ws + o; o += (size_t)G * DFEAT;
    float* cnt  = ws + o; o += (size_t)G;          o = (o + 63) & ~(size_t)63;
    float* zbuf = ws + o; o += (size_t)G * H;
    (void)ws_size; (void)n_in;

    const int BT = 256;
    const int WPB = BT / 32;                 // waves per block
    int nodeBlocks = (N + BT - 1) / BT;
    int edgeBlocks = (E + BT - 1) / BT;
    int edgeWaves  = (E + 1) / 2;            // 2 edges per wave
    int edgeWaveBlocks = (edgeWaves + WPB - 1) / WPB;
    int poolWaves  = (N + 1) / 2;            // 2 nodes per wave
    int poolWaveBlocks = (poolWaves + WPB - 1) / WPB;
    int featBlocks = (N * DFEAT + BT - 1) / BT;

    // 1) degrees with self-loop, then dinv = rsqrt(deg)
    deg_init_kernel<<<nodeBlocks, BT, 0, stream>>>(dinv, N);
    deg_count_kernel<<<edgeBlocks, BT, 0, stream>>>(dst, dinv, E);
    dinv_kernel<<<nodeBlocks, BT, 0, stream>>>(dinv, N);

    // GEMM launch geometry: 1 wave per 16 rows, 4 waves per 128-thread block
    auto gemmBlocks = [](int M) { int w = (M + 15) / 16; return (w + 3) / 4; };

    // ---- layer 1 ----
    gemm_f32_wmma<DFEAT, DFEAT><<<gemmBlocks(N), 128, 0, stream>>>(emb, W1, nullptr, bufA, N, 0);
    hipMemsetAsync(bufB, 0, (size_t)N * DFEAT * sizeof(float), stream);
    edge_scatter_kernel<<<edgeWaveBlocks, BT, 0, stream>>>(src, dst, dinv, bufA, bufB, E);
    finish_kernel<<<featBlocks, BT, 0, stream>>>(bufB, bufA, dinv, b1, bufB, N, /*relu=*/1);

    // ---- layer 2 ----
    gemm_f32_wmma<DFEAT, DFEAT><<<gemmBlocks(N), 128, 0, stream>>>(bufB, W2, nullptr, bufA, N, 0);
    hipMemsetAsync(bufB, 0, (size_t)N * DFEAT * sizeof(float), stream);
    edge_scatter_kernel<<<edgeWaveBlocks, BT, 0, stream>>>(src, dst, dinv, bufA, bufB, E);
    finish_kernel<<<featBlocks, BT, 0, stream>>>(bufB, bufA, dinv, b2, bufB, N, /*relu=*/0);

    // ---- global mean pool ----
    hipMemsetAsync(sums, 0, (size_t)G * DFEAT * sizeof(float), stream);
    hipMemsetAsync(cnt,  0, (size_t)G * sizeof(float), stream);
    pool_accum_kernel<<<poolWaveBlocks, BT, 0, stream>>>(bufB, batch, sums, cnt, N);
    pool_div_kernel<<<(G * DFEAT + BT - 1) / BT, BT, 0, stream>>>(sums, cnt, G);

    // ---- MLP head ----
    if (H == 128) {
        gemm_f32_wmma<DFEAT, 128><<<gemmBlocks(G), 128, 0, stream>>>(sums, mW1, mb1, zbuf, G, /*relu=*/1);
    } else {
        gemm_fallback<<<(G * H + BT - 1) / BT, BT, 0, stream>>>(sums, mW1, mb1, zbuf, G, DFEAT, H, 1);
    }
    mlp2_kernel<<<(G + BT - 1) / BT, BT, 0, stream>>>(zbuf, mW2, mb2, out, G, H);
}